// HungarianMatcher_63204738727936
// MI455X (gfx1250) — compile-verified
//
#include <hip/hip_runtime.h>
#include <hip/hip_bf16.h>
#include <math.h>
#include <stdint.h>

// Problem constants (from reference setup_inputs)
#define B_  16
#define Q_  900
#define T_  128
#define C_  257
#define QC  4                    // q-rows per workgroup (one per wave32)
#define NBLK_PER_B (Q_ / QC)     // 225

// ---- CDNA5 async global->LDS path (probe via __has_builtin) ----
#if defined(__has_builtin)
#  if __has_builtin(__builtin_amdgcn_global_load_async_to_lds_b128)
#    define HAVE_ASYNC_B128 1
#  endif
#  if __has_builtin(__builtin_amdgcn_global_load_async_to_lds_b32)
#    define HAVE_ASYNC_B32 1
#  endif
#  if __has_builtin(__builtin_amdgcn_s_wait_asynccnt)
#    define HAVE_WAIT_ASYNC 1
#  endif
#endif

// Exact parameter types per hipcc diagnostic:
//   b128: (AS1 int-vec16* src, AS3 int-vec16* dst, Ii offset, Ii cpol)
//   b32 : (AS1 int*       src, AS3 int*       dst, Ii offset, Ii cpol)
typedef int v4i __attribute__((vector_size(16)));
typedef __attribute__((address_space(1))) v4i* gv4p;   // pointer to global v4i
typedef __attribute__((address_space(3))) v4i* lv4p;   // pointer to LDS v4i
typedef __attribute__((address_space(1))) int* gi32p;  // pointer to global int
typedef __attribute__((address_space(3))) int* li32p;  // pointer to LDS int

__global__ __launch_bounds__(128) void hm_cost_kernel(
    const float* __restrict__ logits,    // (B,Q,C)
    const float* __restrict__ pboxes,    // (B,Q,4) cxcywh
    const float* __restrict__ tboxes,    // (B,T,4) cxcywh
    const int*   __restrict__ tlabels,   // (B,T)
    float* __restrict__ out)             // (B,Q,T)
{
    __shared__ float4 s_tbox[T_];
    __shared__ int    s_tlab[T_];
    __shared__ float  s_max[QC];
    __shared__ float  s_sum[QC];

    const int blk  = blockIdx.x;
    const int b    = blk / NBLK_PER_B;
    const int q0   = (blk % NBLK_PER_B) * QC;
    const int t    = threadIdx.x;   // 0..127 -> target index
    const int wave = t >> 5;        // 0..3   -> q-row owner
    const int lane = t & 31;

    // ---- Phase A: stage this batch's targets into LDS (async, overlaps B) ----
    {
        const float* gb = tboxes  + ((size_t)b * T_ + t) * 4;
        const int*   gl = tlabels +  (size_t)b * T_ + t;
#if defined(HAVE_ASYNC_B128) && defined(HAVE_ASYNC_B32)
        __builtin_amdgcn_global_load_async_to_lds_b128(
            (gv4p)(uintptr_t)gb, (lv4p)(uintptr_t)&s_tbox[t], 0, 0);
        __builtin_amdgcn_global_load_async_to_lds_b32(
            (gi32p)(uintptr_t)gl, (li32p)(uintptr_t)&s_tlab[t], 0, 0);
#else
        s_tbox[t] = *(const float4*)gb;
        s_tlab[t] = *gl;
#endif
    }

    // ---- Phase B: softmax stats, one wave per q-row (wave32 reductions) ----
    {
        const float* row = logits + ((size_t)b * Q_ + (q0 + wave)) * (size_t)C_;
        float m = -INFINITY;
        for (int c = lane; c < C_; c += 32) m = fmaxf(m, row[c]);
        for (int off = 16; off > 0; off >>= 1) m = fmaxf(m, __shfl_xor(m, off, 32));
        float s = 0.0f;
        for (int c = lane; c < C_; c += 32) s += expf(row[c] - m);
        for (int off = 16; off > 0; off >>= 1) s += __shfl_xor(s, off, 32);
        if (lane == 0) { s_max[wave] = m; s_sum[wave] = s; }
    }

    // ---- sync: async LDS writes + per-wave stats visible to whole WG ----
#if defined(HAVE_ASYNC_B128) && defined(HAVE_ASYNC_B32)
#  if defined(HAVE_WAIT_ASYNC)
    __builtin_amdgcn_s_wait_asynccnt(0);
#  else
    asm volatile("s_wait_asynccnt 0" ::: "memory");
#  endif
#endif
    __syncthreads();

    // ---- Phase C: thread t vs q0..q0+3 ----
    const float4 tb  = s_tbox[t];
    const int    lab = s_tlab[t];
    const float tx0 = tb.x - 0.5f * tb.z, ty0 = tb.y - 0.5f * tb.w;
    const float tx1 = tb.x + 0.5f * tb.z, ty1 = tb.y + 0.5f * tb.w;
    const float area_t = (tx1 - tx0) * (ty1 - ty0);

    // prefetch first gather address (emits global_prefetch_b8 on gfx1250)
    __builtin_prefetch(&logits[((size_t)b * Q_ + q0) * C_ + lab], 0, 1);

#pragma unroll
    for (int i = 0; i < QC; ++i) {
        const int    q  = q0 + i;
        const size_t rq = (size_t)b * Q_ + q;
        if (i + 1 < QC)
            __builtin_prefetch(&logits[(rq + 1) * C_ + lab], 0, 1);

        const float4 pb = *(const float4*)(pboxes + rq * 4);

        // class cost: -softmax(logits)[lab]   (exp(x - max) / sum, as jax.nn.softmax)
        const float x    = logits[rq * C_ + lab];
        const float prob = expf(x - s_max[i]) / s_sum[i];

        // L1 box cost
        const float l1 = fabsf(pb.x - tb.x) + fabsf(pb.y - tb.y)
                       + fabsf(pb.z - tb.z) + fabsf(pb.w - tb.w);

        // GIoU cost
        const float px0 = pb.x - 0.5f * pb.z, py0 = pb.y - 0.5f * pb.w;
        const float px1 = pb.x + 0.5f * pb.z, py1 = pb.y + 0.5f * pb.w;
        const float area_p = (px1 - px0) * (py1 - py0);
        const float iw = fmaxf(fminf(px1, tx1) - fmaxf(px0, tx0), 0.0f);
        const float ih = fmaxf(fminf(py1, ty1) - fmaxf(py0, ty0), 0.0f);
        const float inter = iw * ih;
        const float uni   = area_p + area_t - inter;
        const float iou   = inter / uni;
        const float ew = fmaxf(fmaxf(px1, tx1) - fminf(px0, tx0), 0.0f);
        const float eh = fmaxf(fmaxf(py1, ty1) - fminf(py0, ty0), 0.0f);
        const float ae = ew * eh;
        const float giou = iou - (ae - uni) / ae;

        // cost = 1*(-prob) + 5*L1 + 2*(-giou)
        out[rq * T_ + t] = 5.0f * l1 - prob - 2.0f * giou;
    }
}

extern "C" void kernel_launch(void* const* d_in, const int* in_sizes, int n_in,
                              void* d_out, int out_size, void* d_ws, size_t ws_size,
                              hipStream_t stream) {
    const float* logits  = (const float*)d_in[0];   // pred_logits (16,900,257)
    const float* pboxes  = (const float*)d_in[1];   // pred_boxes  (16,900,4)
    const float* tboxes  = (const float*)d_in[2];   // tgt_boxes   (16,128,4)
    const int*   tlabels = (const int*)  d_in[3];   // tgt_labels  (16,128)
    float*       out     = (float*)d_out;           // (16,900,128)

    (void)in_sizes; (void)n_in; (void)out_size; (void)d_ws; (void)ws_size;

    hm_cost_kernel<<<dim3(B_ * NBLK_PER_B), dim3(128), 0, stream>>>(
        logits, pboxes, tboxes, tlabels, out);
}